// FinefyLatticeModule_25400436588642
// MI455X (gfx1250) — compile-verified
//
#include <hip/hip_runtime.h>

typedef __attribute__((ext_vector_type(2))) float v2f;
typedef __attribute__((ext_vector_type(8))) float v8f;

#define N_FINE_C        1000000
#define VAL_DIM_C       32
#define FILTER_EXTENT_C 9
#define NR_FILTERS_C    64
#define K_TOTAL_C       (FILTER_EXTENT_C * VAL_DIM_C)   // 288
#define ROWS_PER_WAVE   32
#define WAVES_PER_BLOCK 8
#define ROWS_PER_BLOCK  (ROWS_PER_WAVE * WAVES_PER_BLOCK) // 256
#define WLDS_FLOATS     (K_TOTAL_C * NR_FILTERS_C)        // 18432 floats = 72 KB

__launch_bounds__(256, 1)
__global__ void finefy_wmma_kernel(const float* __restrict__ coarse,
                                   const int*   __restrict__ nbr,
                                   const float* __restrict__ weight,
                                   float*       __restrict__ out) {
    // Weight staged pair-interleaved: element (k, n) -> wlds[(k>>1)*128 + n*2 + (k&1)]
    // so a B fragment (two consecutive K rows for one column) is one aligned float2.
    extern __shared__ float wlds[];

    const int tid    = threadIdx.x;
    const int lane   = tid & 31;
    const int wave   = tid >> 5;
    const int lane16 = lane & 15;
    const int hi     = lane >> 4;   // 0: K-pair {0,1}, 1: K-pair {2,3} within a 4-step

    // ---- stage weight into LDS (one pass, coalesced reads) ----
    for (int e = tid; e < WLDS_FLOATS; e += 256) {
        int k = e >> 6;     // e = k*64 + n
        int n = e & 63;
        wlds[((k >> 1) << 7) + (n << 1) + (k & 1)] = weight[e];
    }
    __syncthreads();

    // ---- per-wave output tile: rows [rowBase, rowBase+32), cols [0, 64) ----
    const long long rowBase = (long long)blockIdx.x * ROWS_PER_BLOCK
                            + (long long)wave * ROWS_PER_WAVE;

    long long rr0 = rowBase + lane16;         // M-block 0 row for this lane
    long long rr1 = rowBase + 16 + lane16;    // M-block 1 row for this lane
    const int r0 = (int)(rr0 < (N_FINE_C - 1) ? rr0 : (N_FINE_C - 1));
    const int r1 = (int)(rr1 < (N_FINE_C - 1) ? rr1 : (N_FINE_C - 1));

    // Gather base offsets (floats) for the 9 coarse neighbors of each row.
    int base0[FILTER_EXTENT_C], base1[FILTER_EXTENT_C];
#pragma unroll
    for (int j = 0; j < FILTER_EXTENT_C; ++j) {
        base0[j] = nbr[r0 * FILTER_EXTENT_C + j] * VAL_DIM_C;
        base1[j] = nbr[r1 * FILTER_EXTENT_C + j] * VAL_DIM_C;
    }

    const v8f vzero = {0.f, 0.f, 0.f, 0.f, 0.f, 0.f, 0.f, 0.f};
    v8f acc[2][4];
#pragma unroll
    for (int m = 0; m < 2; ++m)
#pragma unroll
        for (int t = 0; t < 4; ++t)
            acc[m][t] = vzero;

    // ---- K loop: 9 neighbors x 8 sub-steps of K=4 -> 72 steps, 8 WMMA each ----
#pragma unroll
    for (int j = 0; j < FILTER_EXTENT_C; ++j) {
#pragma unroll
        for (int s = 0; s < 8; ++s) {
            const int koff = 4 * s + 2 * hi;            // offset inside 32-float block
            // A fragments: 16x4 f32, lane L -> (M = L%16, K = kbase + 2*(L/16))
            v2f a0 = *(const v2f*)(coarse + base0[j] + koff);
            v2f a1 = *(const v2f*)(coarse + base1[j] + koff);
            // B fragments from LDS, pair index p = K/2 (per-lane)
            const int p = j * 16 + 2 * s + hi;
            const float* bp = wlds + (p << 7) + (lane16 << 1);
#pragma unroll
            for (int t = 0; t < 4; ++t) {
                v2f b = *(const v2f*)(bp + t * 32);
                acc[0][t] = __builtin_amdgcn_wmma_f32_16x16x4_f32(
                    false, a0, false, b, (short)0, acc[0][t], false, false);
                acc[1][t] = __builtin_amdgcn_wmma_f32_16x16x4_f32(
                    false, a1, false, b, (short)0, acc[1][t], false, false);
            }
        }
    }

    // ---- store: C/D layout, VGPR r of lane L -> (M = r + 8*(L/16), N = L%16) ----
#pragma unroll
    for (int m = 0; m < 2; ++m) {
#pragma unroll
        for (int r = 0; r < 8; ++r) {
            const long long row = rowBase + m * 16 + 8 * hi + r;
            if (row < N_FINE_C) {
                float* orow = out + row * NR_FILTERS_C + lane16;
#pragma unroll
                for (int t = 0; t < 4; ++t)
                    orow[t * 16] = acc[m][t][r];
            }
        }
    }
}

extern "C" void kernel_launch(void* const* d_in, const int* in_sizes, int n_in,
                              void* d_out, int out_size, void* d_ws, size_t ws_size,
                              hipStream_t stream) {
    const float* coarse = (const float*)d_in[0];   // [250000, 32] f32
    const int*   nbr    = (const int*)d_in[1];     // [1000000, 9] i32
    const float* weight = (const float*)d_in[2];   // [288, 64] f32
    float*       out    = (float*)d_out;           // [1000000, 64] f32

    const int blocks = (N_FINE_C + ROWS_PER_BLOCK - 1) / ROWS_PER_BLOCK; // 3907
    const size_t lds_bytes = WLDS_FLOATS * sizeof(float);                // 73728

    finefy_wmma_kernel<<<blocks, 256, lds_bytes, stream>>>(coarse, nbr, weight, out);
}